// Fusion_Correlation_69389491634611
// MI455X (gfx1250) — compile-verified
//
#include <hip/hip_runtime.h>
#include <hip/hip_bf16.h>

typedef __attribute__((ext_vector_type(2))) float v2f;
typedef __attribute__((ext_vector_type(8))) float v8f;

#define BN_EPS 1e-5f

// ---------------------------------------------------------------------------
// Depthwise 3x3 conv (pad 1, groups=C) + bias.
// One block per (b,c) plane; plane staged in LDS; one thread per pixel.
// ---------------------------------------------------------------------------
template <int H, int W>
__global__ void dw3x3_kernel(const float* __restrict__ in,
                             const float* __restrict__ w,
                             const float* __restrict__ bias,
                             float* __restrict__ out, int C) {
  constexpr int HW = H * W;
  __shared__ float tile[HW];
  const int bc = blockIdx.x;  // b*C + c
  const int c = bc % C;
  const long long base = (long long)bc * HW;
  const int t = threadIdx.x;
  if (t < HW) tile[t] = in[base + t];
  __syncthreads();
  if (t >= HW) return;
  const int y = t / W, x = t % W;
  const float* wp = w + c * 9;
  float acc = bias[c];
#pragma unroll
  for (int i = 0; i < 3; ++i) {
    const int yy = y + i - 1;
    if (yy < 0 || yy >= H) continue;
#pragma unroll
    for (int j = 0; j < 3; ++j) {
      const int xx = x + j - 1;
      if (xx < 0 || xx >= W) continue;
      acc = fmaf(wp[i * 3 + j], tile[yy * W + xx], acc);
    }
  }
  out[base + t] = acc;
}

// ---------------------------------------------------------------------------
// Depthwise dynamic correlation (reference's range(PAD) loops, pad=3):
// zx[b,c,h,w] = sum_{i,j<3} xpad[b,c,h+i,w+j] * zf[b,c,i,j]
// One block per (b,c); xf plane + 3x3 z corner staged in LDS.
// ---------------------------------------------------------------------------
__global__ void corr_kernel(const float* __restrict__ xf,
                            const float* __restrict__ zf,
                            float* __restrict__ out, int C) {
  constexpr int H = 15, W = 15, HW = H * W;
  __shared__ float xs[HW];
  __shared__ float zs[9];
  const int bc = blockIdx.x;
  const long long xbase = (long long)bc * HW;
  const long long zbase = (long long)bc * 49;
  const int t = threadIdx.x;
  if (t < HW) xs[t] = xf[xbase + t];
  if (t < 9) zs[t] = zf[zbase + (t / 3) * 7 + (t % 3)];
  __syncthreads();
  if (t >= HW) return;
  const int y = t / W, x = t % W;
  float acc = 0.f;
#pragma unroll
  for (int i = 0; i < 3; ++i) {
    const int yy = y + i - 3;
    if (yy < 0 || yy >= H) continue;
#pragma unroll
    for (int j = 0; j < 3; ++j) {
      const int xx = x + j - 3;
      if (xx < 0 || xx >= W) continue;
      acc = fmaf(xs[yy * W + xx], zs[i * 3 + j], acc);
    }
  }
  out[xbase + t] = acc;
}

// ---------------------------------------------------------------------------
// Pointwise (1x1) conv as per-image GEMM + bias + BN + ReLU via
// V_WMMA_F32_16X16X4_F32 (full f32 — the workload is HBM-bound at ~3 TFLOP/s
// needed, so no reason to downconvert).
//   act: (B, K, NS)   Wmat: (M, K) row-major   out: (B, M, NS)
// grid.x = B * (M/16); 128 threads = 4 waves; each wave owns N-tiles of 16.
// K, NS compile-time -> all K-step addresses fold into load immediates.
// Out-of-range columns clamp their load address (their GEMM columns are
// never stored), so the hot loop is branch-free.
// ---------------------------------------------------------------------------
template <int K, int NS>
__global__ __launch_bounds__(128) void pw_bn_relu_wmma_kernel(
    const float* __restrict__ act, const float* __restrict__ Wmat,
    const float* __restrict__ pwb, const float* __restrict__ gamma,
    const float* __restrict__ beta, const float* __restrict__ mean,
    const float* __restrict__ var, float* __restrict__ out, int M) {
  __shared__ float wtile[16 * K];   // <= 16 KB
  __shared__ float sscale[16], sshift[16];

  const int mtiles = M >> 4;
  const int b = blockIdx.x / mtiles;
  const int co0 = (blockIdx.x % mtiles) << 4;
  const int tid = threadIdx.x;

  // Stage 16 contiguous weight rows into LDS (reused by all waves/N-tiles),
  // and fold bias+BN into per-channel scale/shift.
  const float* wsrc = Wmat + (long long)co0 * K;
  for (int i = tid; i < 16 * K; i += 128) wtile[i] = wsrc[i];
  if (tid < 16) {
    const int co = co0 + tid;
    const float sc = gamma[co] * rsqrtf(var[co] + BN_EPS);
    sscale[tid] = sc;
    sshift[tid] = (pwb[co] - mean[co]) * sc + beta[co];
  }
  __syncthreads();

  const int wave = tid >> 5;
  const int lane = tid & 31;
  const int half = lane >> 4;  // 0: K0/K1 + C rows 0-7 ; 1: K2/K3 + C rows 8-15
  const int l16 = lane & 15;

  const float* actb = act + (long long)b * K * NS;
  float* outb = out + (long long)b * M * NS;
  constexpr int ntiles = (NS + 15) / 16;

  for (int nt = wave; nt < ntiles; nt += 4) {
    const int n = (nt << 4) + l16;
    const bool nvalid = (n < NS);
    const int nc = nvalid ? n : NS - 1;           // clamp, never stored
    const float* ap = actb + nc + 2 * half * NS;  // B-tile base for this lane
    const float* wp = wtile + l16 * K + 2 * half; // A-tile base (LDS)
    v8f acc = {};
#pragma unroll 8
    for (int k0 = 0; k0 < K; k0 += 4) {
      v2f a, bm;
      a.x = wp[k0];                // ds_load_b64, const offset
      a.y = wp[k0 + 1];
      bm.x = ap[k0 * NS];          // global_load_b32, const offsets
      bm.y = ap[k0 * NS + NS];
      acc = __builtin_amdgcn_wmma_f32_16x16x4_f32(
          /*neg_a=*/false, a, /*neg_b=*/false, bm,
          /*c_mod=*/(short)0, acc, /*reuse_a=*/false, /*reuse_b=*/false);
    }
    if (nvalid) {
      float* op = outb + (long long)(co0 + 8 * half) * NS + n;
#pragma unroll
      for (int r = 0; r < 8; ++r) {
        const int cr = r + 8 * half;
        const float val = fmaf(acc[r], sscale[cr], sshift[cr]);
        op[r * NS] = fmaxf(val, 0.f);
      }
    }
  }
}

// ---------------------------------------------------------------------------
// Host-side orchestration
// ---------------------------------------------------------------------------
extern "C" void kernel_launch(void* const* d_in, const int* in_sizes, int n_in,
                              void* d_out, int out_size, void* d_ws, size_t ws_size,
                              hipStream_t stream) {
  (void)in_sizes; (void)n_in; (void)out_size; (void)ws_size;

  const int B = 512, CIN = 256, C = 128, COUT = 256;
  const int HWZ = 49, HWX = 225;

  const float* z = (const float*)d_in[0];
  const float* x = (const float*)d_in[1];

  // flattened param dict order: pre_z[0], pre_z[1], pre_x[0], pre_x[1],
  // post_a, post_b (block_k3: dw_w,dw_b,pw_w,pw_b,g,b,m,v), post_c (w,bb,g,b,m,v)
  const int PZ0 = 2, PZ1 = 10, PX0 = 18, PX1 = 26, PA = 34, PB = 42, PC1 = 50;
#define P(base, off) ((const float*)d_in[(base) + (off)])

  float* ws = (float*)d_ws;
  float* BUF_A = ws;                                   // B*256*225 (dw scratch)
  float* BUF_B = BUF_A + (long long)B * CIN * HWX;     // B*128*225
  float* BUF_C = BUF_B + (long long)B * C * HWX;       // B*128*225
  float* BUF_Z0 = BUF_C + (long long)B * C * HWX;      // B*128*49
  float* BUF_Z1 = BUF_Z0 + (long long)B * C * HWZ;     // B*128*49

  const dim3 T256(256), T128(128), T64(64);

  // ---- x branch: pre_x[0] ----
  dw3x3_kernel<15, 15><<<B * CIN, T256, 0, stream>>>(x, P(PX0, 0), P(PX0, 1),
                                                     BUF_A, CIN);
  pw_bn_relu_wmma_kernel<256, 225><<<B * (C / 16), T128, 0, stream>>>(
      BUF_A, P(PX0, 2), P(PX0, 3), P(PX0, 4), P(PX0, 5), P(PX0, 6), P(PX0, 7),
      BUF_B, C);
  // ---- pre_x[1] ----
  dw3x3_kernel<15, 15><<<B * C, T256, 0, stream>>>(BUF_B, P(PX1, 0), P(PX1, 1),
                                                   BUF_A, C);
  pw_bn_relu_wmma_kernel<128, 225><<<B * (C / 16), T128, 0, stream>>>(
      BUF_A, P(PX1, 2), P(PX1, 3), P(PX1, 4), P(PX1, 5), P(PX1, 6), P(PX1, 7),
      BUF_C, C);                                       // BUF_C = xf

  // ---- z branch: pre_z[0] ----
  dw3x3_kernel<7, 7><<<B * CIN, T64, 0, stream>>>(z, P(PZ0, 0), P(PZ0, 1),
                                                  BUF_A, CIN);
  pw_bn_relu_wmma_kernel<256, 49><<<B * (C / 16), T128, 0, stream>>>(
      BUF_A, P(PZ0, 2), P(PZ0, 3), P(PZ0, 4), P(PZ0, 5), P(PZ0, 6), P(PZ0, 7),
      BUF_Z0, C);
  // ---- pre_z[1] ----
  dw3x3_kernel<7, 7><<<B * C, T64, 0, stream>>>(BUF_Z0, P(PZ1, 0), P(PZ1, 1),
                                                BUF_A, C);
  pw_bn_relu_wmma_kernel<128, 49><<<B * (C / 16), T128, 0, stream>>>(
      BUF_A, P(PZ1, 2), P(PZ1, 3), P(PZ1, 4), P(PZ1, 5), P(PZ1, 6), P(PZ1, 7),
      BUF_Z1, C);                                      // BUF_Z1 = zf

  // ---- depthwise correlation ----
  corr_kernel<<<B * C, T256, 0, stream>>>(BUF_C, BUF_Z1, BUF_B, C);

  // ---- post_a ----
  dw3x3_kernel<15, 15><<<B * C, T256, 0, stream>>>(BUF_B, P(PA, 0), P(PA, 1),
                                                   BUF_A, C);
  pw_bn_relu_wmma_kernel<128, 225><<<B * (C / 16), T128, 0, stream>>>(
      BUF_A, P(PA, 2), P(PA, 3), P(PA, 4), P(PA, 5), P(PA, 6), P(PA, 7),
      BUF_B, C);
  // ---- post_b ----
  dw3x3_kernel<15, 15><<<B * C, T256, 0, stream>>>(BUF_B, P(PB, 0), P(PB, 1),
                                                   BUF_A, C);
  pw_bn_relu_wmma_kernel<128, 225><<<B * (C / 16), T128, 0, stream>>>(
      BUF_A, P(PB, 2), P(PB, 3), P(PB, 4), P(PB, 5), P(PB, 6), P(PB, 7),
      BUF_C, C);
  // ---- post_c (1x1, 128->256) straight into d_out ----
  pw_bn_relu_wmma_kernel<128, 225><<<B * (COUT / 16), T128, 0, stream>>>(
      BUF_C, P(PC1, 0), P(PC1, 1), P(PC1, 2), P(PC1, 3), P(PC1, 4), P(PC1, 5),
      (float*)d_out, COUT);
#undef P
}